// beam_decoder_6914897347297
// MI455X (gfx1250) — compile-verified
//
#include <hip/hip_runtime.h>
#include <hip/hip_bf16.h>
#include <math.h>

#define VOCAB   50257
#define MAX_OOV 100
#define VEXT    (VOCAB + MAX_OOV)       // 50357
#define VEXT_PAD 50368                  // pad to multiple of 64
#define SEQ     2048
#define BATCH   640
#define EMB     256
#define ZDIM    128
#define ENC2    1024
#define INTER   512
#define TOPK    15
#define NEGBIG  (-1e9f)
#define EPSV    (1e-9f)

typedef float v2f __attribute__((ext_vector_type(2)));
typedef float v8f __attribute__((ext_vector_type(8)));

// ---------------------------------------------------------------------------
// Kernel 1: prob_ptr = sigmoid( tanh(concat @ W_inter + b_inter) @ w_copy + b_copy )
// WMMA f32 16x16x4, wave32: one wave owns a 16(M)x16(N) tile of the hidden
// layer. The 1536-wide concat K-loop is split into 4 straight-line segment
// loops (one per source tensor) so there is no per-iteration branching.
// ---------------------------------------------------------------------------
__device__ __forceinline__ v8f mma_seg(const float* __restrict__ src, int stride,
                                       int kbase, int klen,       // segment [kbase, kbase+klen)
                                       const float* __restrict__ W,
                                       int m, int n, int hi, v8f c)
{
#pragma unroll 4
    for (int k = 0; k < klen; k += 4) {
        const int ka = kbase + k + 2 * hi;         // global K index for this lane half
        v2f a, b;
        a.x = src[m * stride + (k + 2 * hi)];
        a.y = src[m * stride + (k + 2 * hi) + 1];
        b.x = W[(size_t)ka * INTER + n];
        b.y = W[(size_t)(ka + 1) * INTER + n];
        c = __builtin_amdgcn_wmma_f32_16x16x4_f32(
                /*neg_a=*/false, a, /*neg_b=*/false, b,
                /*c_mod=*/(short)0, c, /*reuse_a=*/false, /*reuse_b=*/false);
    }
    return c;
}

__global__ void copy_gate_wmma(const float* __restrict__ input_embs,   // [640,256]
                               const float* __restrict__ extra_feat,   // [640,128]
                               const float* __restrict__ context_vec,  // [640,1024]
                               const float* __restrict__ zvec,         // [640,128]
                               const float* __restrict__ W_inter,      // [1536,512]
                               const float* __restrict__ b_inter,      // [512]
                               const float* __restrict__ w_copy,       // [512]
                               const float* __restrict__ b_copy,       // [1]
                               float* __restrict__ prob_ptr)           // [640]
{
    __shared__ float rowacc[16];
    const int tid  = threadIdx.x;
    const int lane = tid & 31;
    const int wave = tid >> 5;          // 0..3
    const int hi   = lane >> 4;         // 0/1
    const int mn   = lane & 15;
    const int row0 = blockIdx.x * 16;
    const int m    = row0 + mn;

    if (tid < 16) rowacc[tid] = 0.0f;
    __syncthreads();

    // 32 N-tiles of 16 cols; 4 waves -> 8 tiles each
    for (int nt = wave; nt < INTER / 16; nt += 4) {
        const int n = nt * 16 + mn;
        v8f c = {};
        c = mma_seg(input_embs,  EMB,  0,                 EMB,  W_inter, m, n, hi, c);
        c = mma_seg(extra_feat,  ZDIM, EMB,               ZDIM, W_inter, m, n, hi, c);
        c = mma_seg(context_vec, ENC2, EMB + ZDIM,        ENC2, W_inter, m, n, hi, c);
        c = mma_seg(zvec,        ZDIM, EMB + ZDIM + ENC2, ZDIM, W_inter, m, n, hi, c);

        const float bi = b_inter[n];
        const float wc = w_copy[n];
#pragma unroll
        for (int i = 0; i < 8; ++i) {
            const int mrow = i + 8 * hi;    // C layout: VGPR i -> M=i (lanes 0-15) / M=i+8 (lanes 16-31)
            const float h = tanhf(c[i] + bi);
            atomicAdd(&rowacc[mrow], h * wc);
        }
    }
    __syncthreads();
    if (tid < 16) {
        const float v = rowacc[tid] + b_copy[0];
        prob_ptr[row0 + tid] = 1.0f / (1.0f + expf(-v));
    }
}

// ---------------------------------------------------------------------------
// Kernel 2: per-row fused attn-softmax + scatter (201KB accumulator in the
// 320KB CDNA5 WGP LDS) + vocab softmax + log + top-15.
// One 1024-thread workgroup (32 waves) per row; 640 workgroups.
// ---------------------------------------------------------------------------
__global__ void fused_dist_topk(const float* __restrict__ vocab_logits, // [640,50257]
                                const float* __restrict__ attn_logits,  // [640,2048]
                                const float* __restrict__ group_mask,   // [640,2048]
                                const int*   __restrict__ group_src,    // [640,2048]
                                const float* __restrict__ prob_ptr,     // [640]
                                float* __restrict__ out)                // [2*640*15]
{
    extern __shared__ char smem[];
    float* ext  = (float*)smem;                 // VEXT_PAD floats (scatter accum)
    float* redA = ext  + VEXT_PAD;              // 1024
    float* redB = redA + 1024;                  // 1024
    int*   redI = (int*)(redB + 1024);          // 1024

    const int row = blockIdx.x;
    const int tid = threadIdx.x;
    const float pp   = prob_ptr[row];
    const float pgen = 1.0f - pp;

    const float* vl = vocab_logits + (size_t)row * VOCAB;
    const float* al = attn_logits  + (size_t)row * SEQ;
    const float* gm = group_mask   + (size_t)row * SEQ;
    const int*   gs = group_src    + (size_t)row * SEQ;

    // zero the extended-dist accumulator
    for (int i = tid; i < VEXT_PAD; i += 1024) ext[i] = 0.0f;

    // ---- attention softmax: masked max ----
    float amax = -INFINITY;
    for (int s = tid; s < SEQ; s += 1024) {
        const float x = (gm[s] > 0.0f) ? al[s] : NEGBIG;
        amax = fmaxf(amax, x);
    }
    __syncthreads();                    // ext zeroing done before redA reuse
    redA[tid] = amax;
    __syncthreads();
    for (int off = 512; off; off >>= 1) {
        if (tid < off) { amax = fmaxf(amax, redA[tid + off]); redA[tid] = amax; }
        __syncthreads();
    }
    const float AM = redA[0];
    __syncthreads();

    // ---- attention softmax: sum of exp ----
    float asum = 0.0f;
    for (int s = tid; s < SEQ; s += 1024) {
        const float x = (gm[s] > 0.0f) ? al[s] : NEGBIG;
        asum += expf(x - AM);
    }
    redA[tid] = asum;
    __syncthreads();
    for (int off = 512; off; off >>= 1) {
        if (tid < off) { asum += redA[tid + off]; redA[tid] = asum; }
        __syncthreads();
    }
    const float ascale = pp / redA[0];
    __syncthreads();

    // ---- scatter prob_ptr * attn_dist into LDS (ds_add_f32 atomics) ----
    for (int s = tid; s < SEQ; s += 1024) {
        const float x  = (gm[s] > 0.0f) ? al[s] : NEGBIG;
        const float wa = expf(x - AM) * ascale;
        atomicAdd(&ext[gs[s]], wa);
    }

    // ---- vocab softmax pass 1: online (max, sumexp) per thread ----
    float m = -INFINITY, sx = 0.0f;
    for (int id = tid; id < VOCAB; id += 1024) {
        const float x = vl[id];
        if (x > m) { sx = sx * expf(m - x) + 1.0f; m = x; }
        else       { sx += expf(x - m); }
    }
    __syncthreads();                    // scatter complete before redA/redB reuse
    redA[tid] = m; redB[tid] = sx;
    __syncthreads();
    for (int off = 512; off; off >>= 1) {
        if (tid < off) {
            const float m2 = redA[tid + off], s2 = redB[tid + off];
            const float M  = fmaxf(m, m2);
            sx = sx * expf(m - M) + s2 * expf(m2 - M);
            m  = M;
            redA[tid] = m; redB[tid] = sx;
        }
        __syncthreads();
    }
    const float VM   = redA[0];
    const float invS = pgen / redB[0];  // fold p_gen into the normalizer
    __syncthreads();

    // ---- pass 2: final dist + log + per-thread top-15 ----
    float tv[TOPK];
    int   ti[TOPK];
#pragma unroll
    for (int i = 0; i < TOPK; ++i) { tv[i] = -INFINITY; ti[i] = 0; }

    for (int id = tid; id < VEXT; id += 1024) {
        float v = ext[id];
        if (id < VOCAB) v += expf(vl[id] - VM) * invS;
        const float lp = logf(v + EPSV);
        if (lp > tv[TOPK - 1]) {
            int j = TOPK - 1;
            while (j > 0 && tv[j - 1] < lp) { tv[j] = tv[j - 1]; ti[j] = ti[j - 1]; --j; }
            tv[j] = lp; ti[j] = id;
        }
    }

    // ---- block merge: 15 rounds of arg-max over per-thread heads ----
    int head = 0;
    float* probs_out = out + (size_t)row * TOPK;
    float* ids_out   = out + (size_t)BATCH * TOPK + (size_t)row * TOPK;
    for (int r = 0; r < TOPK; ++r) {
        redA[tid] = (head < TOPK) ? tv[head] : -INFINITY;
        redI[tid] = tid;
        __syncthreads();
        for (int off = 512; off; off >>= 1) {
            if (tid < off && redA[tid + off] > redA[tid]) {
                redA[tid] = redA[tid + off];
                redI[tid] = redI[tid + off];
            }
            __syncthreads();
        }
        const int   wt = redI[0];
        const float wv = redA[0];
        __syncthreads();
        if (tid == wt) {
            probs_out[r] = wv;
            ids_out[r]   = (float)ti[head];
            ++head;
        }
    }
}

// ---------------------------------------------------------------------------
extern "C" void kernel_launch(void* const* d_in, const int* in_sizes, int n_in,
                              void* d_out, int out_size, void* d_ws, size_t ws_size,
                              hipStream_t stream) {
    (void)in_sizes; (void)n_in; (void)out_size; (void)ws_size;
    const float* input_embs   = (const float*)d_in[0];
    const float* extra_feat   = (const float*)d_in[1];
    const float* context_vec  = (const float*)d_in[2];
    const float* zvec         = (const float*)d_in[3];
    const float* vocab_logits = (const float*)d_in[4];
    const float* attn_logits  = (const float*)d_in[5];
    const float* group_mask   = (const float*)d_in[6];
    const float* W_inter      = (const float*)d_in[7];
    const float* b_inter      = (const float*)d_in[8];
    const float* w_copy       = (const float*)d_in[9];
    const float* b_copy       = (const float*)d_in[10];
    const int*   group_src    = (const int*)d_in[11];

    float* prob_ptr = (float*)d_ws;   // 640 floats of scratch
    float* out      = (float*)d_out;  // [640*15 probs | 640*15 ids-as-float]

    copy_gate_wmma<<<BATCH / 16, 128, 0, stream>>>(
        input_embs, extra_feat, context_vec, zvec,
        W_inter, b_inter, w_copy, b_copy, prob_ptr);

    const size_t lds_bytes = (size_t)(VEXT_PAD + 3 * 1024) * sizeof(float); // ~214 KB (<320 KB WGP LDS)
    hipFuncSetAttribute((const void*)fused_dist_topk,
                        hipFuncAttributeMaxDynamicSharedMemorySize,
                        (int)lds_bytes);
    fused_dist_topk<<<BATCH, 1024, lds_bytes, stream>>>(
        vocab_logits, attn_logits, group_mask, group_src, prob_ptr, out);
}